// Network_3745211482391
// MI455X (gfx1250) — compile-verified
//
#include <hip/hip_runtime.h>

#define N 4096

typedef float v2f __attribute__((ext_vector_type(2)));
typedef float v4f __attribute__((ext_vector_type(4)));
typedef float v8f __attribute__((ext_vector_type(8)));

// ---------------- Phase 1: partial GEMV  yin @ (w + alpha*hebb) ----------------
// Grid: JT * IC blocks of 256 threads. Block (jt, ic) reduces rows [ic*TI, ic*TI+TI)
// into columns [jt*TJ, jt*TJ+TJ), each thread owning 4 consecutive columns (v4f
// b128 streaming loads). Partials land in d_ws[ic*N + j] (fully overwritten).
constexpr int TJ = 1024;       // columns per block (256 threads * 4 floats)
constexpr int TI = 64;         // rows per block
constexpr int JT = N / TJ;     // 4 column tiles
constexpr int IC = N / TI;     // 64 row chunks  -> 256 blocks, ws = IC*N*4 = 1 MiB

__global__ void gemv_partial(const float* __restrict__ yin,
                             const float* __restrict__ hebb,
                             const float* __restrict__ w,
                             const float* __restrict__ alpha,
                             float* __restrict__ partial) {
  const int jt  = blockIdx.x % JT;
  const int ic  = blockIdx.x / JT;
  const int tid = threadIdx.x;                 // 0..255
  const int j   = jt * TJ + tid * 4;

  __shared__ float syin[TI];
  if (tid < TI) syin[tid] = yin[ic * TI + tid];
  __syncthreads();

  v4f acc = {0.f, 0.f, 0.f, 0.f};
  const int i0 = ic * TI;
#pragma unroll 4
  for (int ii = 0; ii < TI; ++ii) {
    const float  yv   = syin[ii];
    const size_t base = (size_t)(i0 + ii) * N + j;
    // w and alpha are single-use: stream non-temporally. hebb is re-read by
    // phase 3 and may live in the 192MB L2: keep temporal.
    const v4f wv = __builtin_nontemporal_load((const v4f*)(w + base));
    const v4f av = __builtin_nontemporal_load((const v4f*)(alpha + base));
    const v4f hv = *(const v4f*)(hebb + base);
    acc.x = fmaf(yv, fmaf(av.x, hv.x, wv.x), acc.x);
    acc.y = fmaf(yv, fmaf(av.y, hv.y, wv.y), acc.y);
    acc.z = fmaf(yv, fmaf(av.z, hv.z, wv.z), acc.z);
    acc.w = fmaf(yv, fmaf(av.w, hv.w, wv.w), acc.w);
  }
  *(v4f*)(partial + (size_t)ic * N + j) = acc;
}

// ---------------- Phase 2: reduce partials, tanh + clamp-select ----------------
__global__ void finalize_yout(const float* __restrict__ partial,
                              const float* __restrict__ input,
                              float* __restrict__ yout) {
  const int j = blockIdx.x * blockDim.x + threadIdx.x;   // 4096 threads
  float s = 0.f;
#pragma unroll 8
  for (int ic = 0; ic < IC; ++ic) s += partial[(size_t)ic * N + j];
  const float inj = input[j];
  const float t   = tanhf(s);
  yout[j] = (inj != 0.0f) ? inj : t;                     // clamps = (input != 0)
}

// ---------------- Phase 3: hebb' = (1-eta)*hebb + eta*outer(yin,yout) ----------
// One V_WMMA_F32_16X16X4_F32 per 16x16 tile:
//   A (16x4): column K=0 = eta*yin[row tile], K=1..3 = 0
//   B (4x16): row    K=0 = yout[col tile],    K=1..3 = 0
//   C        = (1-eta) * hebb tile
//   D = A*B + C  ->  new hebb tile.
// Wave32, EXEC all-ones (no divergence). 8192 waves = 1024 blocks x 8 waves;
// each wave: one 16-row tile x 8 consecutive 16-col tiles (A reused across tiles).
__global__ void hebb_update(const float* __restrict__ hebb,
                            const float* __restrict__ yin,
                            const float* __restrict__ yout,
                            const float* __restrict__ eta_p,
                            float* __restrict__ hebb_out) {
  const float eta   = eta_p[0];
  const float decay = 1.0f - eta;

  const int lane = threadIdx.x & 31;
  const int wave = threadIdx.x >> 5;
  const int gw   = blockIdx.x * 8 + wave;   // 0..8191
  const int rt   = gw >> 5;                 // row tile   0..255
  const int cs   = gw & 31;                 // col strip  0..31 (8 tiles each)
  const int half = lane >> 4;               // 0: lanes 0-15, 1: lanes 16-31
  const int l15  = lane & 15;

  // A-matrix (16x4 f32, 2 VGPRs): lanes 0-15 carry K=0 (v0) / K=1 (v1);
  // lanes 16-31 carry K=2/K=3. Only K=0 is nonzero.
  v2f a;
  a.x = (half == 0) ? eta * yin[rt * 16 + l15] : 0.0f;
  a.y = 0.0f;

  const int row0 = rt * 16;
#pragma unroll
  for (int t = 0; t < 8; ++t) {
    const int col0 = cs * 128 + t * 16;

    // B-matrix (4x16 f32): row K=0 across lanes 0-15 of v0; everything else 0.
    v2f b;
    b.x = (half == 0) ? yout[col0 + l15] : 0.0f;
    b.y = 0.0f;

    // C/D layout: VGPR r, lane l -> M = r + 8*half, Nc = l&15.
    const size_t cbase = (size_t)(row0 + 8 * half) * N + col0 + l15;
    v8f c;
#pragma unroll
    for (int r = 0; r < 8; ++r)
      c[r] = decay * hebb[cbase + (size_t)r * N];

    v8f d = __builtin_amdgcn_wmma_f32_16x16x4_f32(
        /*neg_a=*/false, a, /*neg_b=*/false, b,
        /*c_mod=*/(short)0, c, /*reuse_a=*/false, /*reuse_b=*/false);

#pragma unroll
    for (int r = 0; r < 8; ++r)
      __builtin_nontemporal_store(d[r], hebb_out + cbase + (size_t)r * N);
  }
}

extern "C" void kernel_launch(void* const* d_in, const int* in_sizes, int n_in,
                              void* d_out, int out_size, void* d_ws, size_t ws_size,
                              hipStream_t stream) {
  const float* input = (const float*)d_in[0];   // [1, N]
  const float* yin   = (const float*)d_in[1];   // [1, N]
  const float* hebb  = (const float*)d_in[2];   // [N, N]
  const float* w     = (const float*)d_in[3];   // [N, N]
  const float* alpha = (const float*)d_in[4];   // [N, N]
  const float* eta   = (const float*)d_in[5];   // [1]

  float* yout     = (float*)d_out;              // first N floats
  float* hebb_out = (float*)d_out + N;          // next N*N floats
  float* partial  = (float*)d_ws;               // IC*N floats = 1 MiB, overwritten

  gemv_partial <<<JT * IC, 256, 0, stream>>>(yin, hebb, w, alpha, partial);
  finalize_yout<<<N / 256, 256, 0, stream>>>(partial, input, yout);
  hebb_update  <<<1024,    256, 0, stream>>>(hebb, yin, yout, eta, hebb_out);
}